// ALiBiPositionalBias_24489903521900
// MI455X (gfx1250) — compile-verified
//
#include <hip/hip_runtime.h>

// ---------------------------------------------------------------------------
// ALiBi positional bias with contextual heads, MI455X (gfx1250, wave32).
//
// out[h,i,j] (16 x S x S fp32, S=2048):
//   heads 4..15 : cross ? cab[h] : slope_h * -|j - i - offset|
//   heads 0..3  : cross ? cab[h] : slope_h * c_pos[i,j]
// where c_bias = -sigmoid(q.k^T / 8) and, using a full-row cumsum cum[]:
//   d = i+offset;  c_pos = j>d ? cum[j]-cum[d] : j<d ? cum[d-1]-cum[j-1] : 0
//
// Store-bound: 256 MB out @ 23.3 TB/s ~ 11 us. GEMM (2.1 GFLOP fp32) done
// with chained V_WMMA_F32_16X16X4_F32 so it stays far off the critical path.
// ---------------------------------------------------------------------------

typedef __attribute__((ext_vector_type(2))) float v2f;
typedef __attribute__((ext_vector_type(8))) float v8f;

#define TOTAL_HEADS 16
#define CTX_HEADS   4
#define MAX_S       2048
#define WAVES       4
#define THREADS     (WAVES * 32)
#define ROW_STRIDE  2080          // 2048 + 32 pad floats (1 per 64) -> conflict-free scan

__device__ __forceinline__ int padj(int j) { return j + (j >> 6); }

__device__ __forceinline__ float slope_of(int h) {
    // TOTAL_HEADS = 16 (power of two): slope[h] = 2^(-(h+1)/2)
    return exp2f(-0.5f * (float)(h + 1));
}

// ============================ contextual heads =============================
__global__ void __launch_bounds__(THREADS)
ctx_heads_kernel(const float* __restrict__ q, const float* __restrict__ k,
                 const float* __restrict__ cab,
                 const int* __restrict__ off_p, const int* __restrict__ pre_p,
                 float* __restrict__ out, int S)
{
    __shared__ float lds[16 * ROW_STRIDE];   // 16 rows of c_bias -> in-place cumsum

    const int h    = blockIdx.y;             // 0..3
    const int row0 = blockIdx.x * 16;        // 16-row tile of q
    const int tid  = threadIdx.x;
    const int lane = tid & 31;
    const int wid  = tid >> 5;
    const int offset = off_p[0];
    const int prefix = pre_p[0];

    const float* qh = q + (size_t)h * S * 64;
    const float* kh = k + (size_t)h * S * 64;

    // ---- A fragments (f32 16x16x4 layout): lane l -> row l%16, K-pair 2*(l/16)
    const int m  = lane & 15;
    const int kp = (lane >> 4) * 2;
    v2f a[16];
    {
        const float* qrow = qh + (size_t)(row0 + m) * 64 + kp;
        #pragma unroll
        for (int f = 0; f < 16; ++f) { a[f].x = qrow[f * 4]; a[f].y = qrow[f * 4 + 1]; }
    }

    // ---- GEMM phase: each wave takes every WAVES-th 16-column tile
    const int ntiles = S >> 4;
    for (int ct = wid; ct < ntiles; ct += WAVES) {
        const int c0 = ct * 16;
        v2f bf[16];                                  // B = k^T, same per-lane layout
        const float* krow = kh + (size_t)(c0 + m) * 64 + kp;
        #pragma unroll
        for (int f = 0; f < 16; ++f) { bf[f].x = krow[f * 4]; bf[f].y = krow[f * 4 + 1]; }

        v8f c = {0.f, 0.f, 0.f, 0.f, 0.f, 0.f, 0.f, 0.f};
        #pragma unroll
        for (int f = 0; f < 16; ++f) {
            c = __builtin_amdgcn_wmma_f32_16x16x4_f32(
                    false, a[f], false, bf[f], (short)0, c, false, false);
        }

        // C layout: lane -> N=lane%16; vgpr v -> M = v + 8*(lane/16)
        const int rbase = (lane >> 4) * 8;
        const int col   = c0 + (lane & 15);
        #pragma unroll
        for (int v = 0; v < 8; ++v) {
            float x  = c[v] * 0.125f;                 // * d^-0.5, d=64
            float cb = -1.0f / (1.0f + __expf(-x));   // -sigmoid
            lds[(rbase + v) * ROW_STRIDE + padj(col)] = cb;
        }
    }
    __syncthreads();

    // ---- Scan phase: 4 rows per wave, full-row inclusive cumsum in place
    const int csz = S >> 5;                           // elements per lane
    for (int r = wid * 4; r < wid * 4 + 4; ++r) {
        float* row = lds + r * ROW_STRIDE;
        const int cb0 = lane * csz;
        float s0 = 0.f;
        for (int t = 0; t < csz; ++t) s0 += row[padj(cb0 + t)];
        float v = s0;                                 // wave inclusive scan of partials
        #pragma unroll
        for (int d = 1; d < 32; d <<= 1) {
            float t2 = __shfl_up(v, d, 32);
            if (lane >= d) v += t2;
        }
        float run = v - s0;                           // exclusive offset for this lane
        for (int t = 0; t < csz; ++t) {
            run += row[padj(cb0 + t)];
            row[padj(cb0 + t)] = run;
        }
    }
    __syncthreads();

    // ---- Output phase
    const float slope  = slope_of(h);
    const float crossb = cab[h];
    for (int r = wid * 4; r < wid * 4 + 4; ++r) {
        const int gi   = row0 + r;
        float* row     = lds + r * ROW_STRIDE;
        const int diag = gi + offset;
        auto cumAt = [&](int x) -> float {
            if (x < 0) return 0.f;
            if (x >= S) x = S - 1;
            return row[padj(x)];
        };
        const float cd    = cumAt(diag);
        const float cdm1  = cumAt(diag - 1);
        const bool rowPre = (gi + offset) < prefix;
        float* orow = out + ((size_t)h * S + gi) * S;
        for (int j = lane; j < S; j += 32) {
            float cpos;
            if (j > diag)      cpos = row[padj(j)] - cd;
            else if (j < diag) cpos = cdm1 - cumAt(j - 1);
            else               cpos = 0.f;
            bool isCross = rowPre ^ (j < prefix);
            orow[j] = isCross ? crossb : slope * cpos;
        }
    }
}

// ============================ plain ALiBi heads ============================
__global__ void __launch_bounds__(256)
plain_heads_kernel(const float* __restrict__ cab,
                   const int* __restrict__ off_p, const int* __restrict__ pre_p,
                   float* __restrict__ out, int S)
{
    const int h  = CTX_HEADS + blockIdx.z;
    const int i  = blockIdx.y;
    const int j0 = (blockIdx.x * blockDim.x + threadIdx.x) * 4;
    if (j0 + 3 >= S + 3) return;                      // grid covers S exactly anyway
    const int offset = off_p[0];
    const int prefix = pre_p[0];
    const float slope  = slope_of(h);
    const float crossb = cab[h];
    const int  d       = i + offset;
    const bool rowPre  = d < prefix;

    float4 v;
    #pragma unroll
    for (int t = 0; t < 4; ++t) {
        int j = j0 + t;
        float val = slope * -fabsf((float)(j - d));
        bool isCross = rowPre ^ (j < prefix);
        (&v.x)[t] = isCross ? crossb : val;
    }
    *(float4*)(out + ((size_t)h * S + i) * S + j0) = v;
}

// ================================ launcher =================================
extern "C" void kernel_launch(void* const* d_in, const int* in_sizes, int n_in,
                              void* d_out, int out_size, void* d_ws, size_t ws_size,
                              hipStream_t stream)
{
    const float* q     = (const float*)d_in[0];
    const float* k     = (const float*)d_in[1];
    const float* cab   = (const float*)d_in[2];
    const int*   off_p = (const int*)d_in[5];
    const int*   pre_p = (const int*)d_in[6];
    float* out = (float*)d_out;

    const int S = in_sizes[0] / (TOTAL_HEADS * 64);   // 2048 for the reference setup

    dim3 gA(S / 16, CTX_HEADS);
    ctx_heads_kernel<<<gA, THREADS, 0, stream>>>(q, k, cab, off_p, pre_p, out, S);

    dim3 gB(S / 1024, S, TOTAL_HEADS - CTX_HEADS);    // 256 thr * 4 floats = 1024 cols
    plain_heads_kernel<<<gB, 256, 0, stream>>>(cab, off_p, pre_p, out, S);
}